// GCN_Conv_50500225466808
// MI455X (gfx1250) — compile-verified
//
#include <hip/hip_runtime.h>
#include <math.h>

typedef float v2f __attribute__((ext_vector_type(2)));
typedef float v8f __attribute__((ext_vector_type(8)));

// ---------------------------------------------------------------------------
// Degree count (in-degree via atomics); self-loop +1 folded into dinv kernel.
// ---------------------------------------------------------------------------
__global__ void deg_kernel(const long long* __restrict__ dst,
                           float* __restrict__ deg, int E) {
    int e = blockIdx.x * blockDim.x + threadIdx.x;
    if (e >= E) return;
    atomicAdd(&deg[dst[e]], 1.0f);
}

__global__ void dinv_kernel(const float* __restrict__ deg,
                            float* __restrict__ dinv, int n) {
    int i = blockIdx.x * blockDim.x + threadIdx.x;
    if (i >= n) return;
    dinv[i] = rsqrtf(deg[i] + 1.0f);   // +1 self loop => deg >= 1
}

// ---------------------------------------------------------------------------
// Pad weights [K x Hsrc] -> [K x 16] (zeros in cols Hsrc..15) for WMMA B tile.
// ---------------------------------------------------------------------------
__global__ void pad_w_kernel(const float* __restrict__ W,
                             float* __restrict__ Wp, int K, int Hsrc) {
    int i = blockIdx.x * blockDim.x + threadIdx.x;
    if (i >= K * 16) return;
    int k = i >> 4, n = i & 15;
    Wp[i] = (n < Hsrc) ? W[k * Hsrc + n] : 0.0f;
}

// ---------------------------------------------------------------------------
// Out[N x 8] = A[N x K] @ Bp[K x 16]  using V_WMMA_F32_16X16X4_F32.
// One wave32 computes one 16x16 tile; K consumed 4 at a time.
// A operand (16x4 f32, 2 VGPRs): lanes 0-15 hold K = k+{0,1}, lanes 16-31
// hold K = k+{2,3}, M = lane&15.  B operand mirrored (rows striped the same
// way).  C/D: VGPR v -> row v (+8 for upper lane half), col = lane&15.
// Bp staged through LDS (max 128x16 f32 = 8KB << 320KB/WGP).
// ---------------------------------------------------------------------------
__global__ void wmma_gemm16_kernel(const float* __restrict__ A, int lda, int K,
                                   const float* __restrict__ Bp,
                                   float* __restrict__ Out, int n) {
    __shared__ float ldsB[128 * 16];
    int t = threadIdx.x;
    for (int i = t; i < K * 16; i += blockDim.x) ldsB[i] = Bp[i];
    __syncthreads();

    int wave = t >> 5;
    int lane = t & 31;
    int tile = blockIdx.x * 8 + wave;     // 8 waves / 256-thread block
    int row0 = tile * 16;
    if (row0 >= n) return;                // whole-wave uniform exit (EXEC stays full)

    int m  = lane & 15;                   // A row within tile / B-D column
    int kb = (lane >> 4) << 1;            // 0 for lanes 0-15, 2 for lanes 16-31

    int row = row0 + m;
    if (row >= n) row = n - 1;            // clamp reads; stores guarded below
    const float* arow = A + (size_t)row * lda;

    v8f c = {};
    for (int kk = 0; kk < K; kk += 4) {
        v2f a, b;
        a.x = arow[kk + kb];
        a.y = arow[kk + kb + 1];
        b.x = ldsB[(kk + kb) * 16 + m];
        b.y = ldsB[(kk + kb + 1) * 16 + m];
        // (neg_a, A, neg_b, B, c_mod, C, reuse_a, reuse_b)
        c = __builtin_amdgcn_wmma_f32_16x16x4_f32(
                false, a, false, b, (short)0, c, false, false);
    }

    int rbase = row0 + ((lane >> 4) << 3);  // +8 rows for upper lane half
#pragma unroll
    for (int v = 0; v < 8; ++v) {
        int r = rbase + v;
        if (m < 8 && r < n) Out[(size_t)r * 8 + m] = c[v];
    }
}

// ---------------------------------------------------------------------------
// Edge aggregation: agg[dst] += h[src] * dinv[src] * dinv[dst].
// 1 thread/edge, two float4 gathers (L2-resident: h is 6.4MB), 8 f32 atomics.
// ---------------------------------------------------------------------------
__global__ void edge_agg_kernel(const long long* __restrict__ src,
                                const long long* __restrict__ dst,
                                const float* __restrict__ dinv,
                                const float* __restrict__ h,
                                float* __restrict__ agg, int E) {
    int e = blockIdx.x * blockDim.x + threadIdx.x;
    if (e >= E) return;
    long long s = src[e], d = dst[e];
    float w = dinv[s] * dinv[d];
    const float4* hp = (const float4*)(h + s * 8);
    float4 h0 = hp[0];
    float4 h1 = hp[1];
    float* ap = agg + d * 8;
    atomicAdd(ap + 0, h0.x * w);
    atomicAdd(ap + 1, h0.y * w);
    atomicAdd(ap + 2, h0.z * w);
    atomicAdd(ap + 3, h0.w * w);
    atomicAdd(ap + 4, h1.x * w);
    atomicAdd(ap + 5, h1.y * w);
    atomicAdd(ap + 6, h1.z * w);
    atomicAdd(ap + 7, h1.w * w);
}

// ---------------------------------------------------------------------------
// Finish a layer: add self-loop term h[i]/deg + bias, ReLU; in place on agg.
// (self-loop norm = dinv[i]*dinv[i] = 1/deg[i])
// ---------------------------------------------------------------------------
__global__ void finish_kernel(const float* __restrict__ h,
                              const float* __restrict__ dinv,
                              const float* __restrict__ bias,
                              float* __restrict__ agg, int n) {
    int i = blockIdx.x * blockDim.x + threadIdx.x;
    if (i >= n * 8) return;
    int node = i >> 3, j = i & 7;
    float di = dinv[node];
    float v = agg[i] + h[i] * di * di + bias[j];
    agg[i] = v > 0.0f ? v : 0.0f;
}

// ---------------------------------------------------------------------------
// Per-graph node-sum readout (batch is sorted int64).
// ---------------------------------------------------------------------------
__global__ void readout_kernel(const long long* __restrict__ batch,
                               const float* __restrict__ h,
                               float* __restrict__ gsum, int n) {
    int i = blockIdx.x * blockDim.x + threadIdx.x;
    if (i >= n) return;
    long long g = batch[i];
    const float4* hp = (const float4*)(h + (size_t)i * 8);
    float4 a = hp[0];
    float4 b = hp[1];
    float* gp = gsum + g * 8;
    atomicAdd(gp + 0, a.x);
    atomicAdd(gp + 1, a.y);
    atomicAdd(gp + 2, a.z);
    atomicAdd(gp + 3, a.w);
    atomicAdd(gp + 4, b.x);
    atomicAdd(gp + 5, b.y);
    atomicAdd(gp + 6, b.z);
    atomicAdd(gp + 7, b.w);
}

__global__ void logits_kernel(const float* __restrict__ gsum,
                              const float* __restrict__ Wl,
                              const float* __restrict__ bl,
                              float* __restrict__ out, int G) {
    int g = blockIdx.x * blockDim.x + threadIdx.x;
    if (g >= G) return;
    float acc = bl[0];
#pragma unroll
    for (int j = 0; j < 8; ++j) acc += gsum[g * 8 + j] * Wl[j];
    out[g] = 1.0f / (1.0f + expf(-acc));
}

// ---------------------------------------------------------------------------
// Launcher
// ---------------------------------------------------------------------------
extern "C" void kernel_launch(void* const* d_in, const int* in_sizes, int n_in,
                              void* d_out, int out_size, void* d_ws, size_t ws_size,
                              hipStream_t stream) {
    const float*     x     = (const float*)d_in[0];
    const float*     W1    = (const float*)d_in[1];
    const float*     b1    = (const float*)d_in[2];
    const float*     W2    = (const float*)d_in[3];
    const float*     b2    = (const float*)d_in[4];
    const float*     Wl    = (const float*)d_in[5];
    const float*     bl    = (const float*)d_in[6];
    const long long* ei    = (const long long*)d_in[7];   // int64 [2, E]
    const long long* batch = (const long long*)d_in[8];   // int64 [N], sorted

    const int F = 128, H = 8, G = 1024;
    int N = in_sizes[0] / F;
    int E = in_sizes[7] / 2;
    const long long* src = ei;
    const long long* dst = ei + E;

    // Workspace carve (256B aligned)
    char* ws = (char*)d_ws;
    size_t off = 0;
    auto carve = [&](size_t bytes) {
        char* p = ws + off;
        off = (off + bytes + 255) & ~(size_t)255;
        return p;
    };
    float* deg  = (float*)carve((size_t)N * 4);
    float* dinv = (float*)carve((size_t)N * 4);
    float* h    = (float*)carve((size_t)N * 8 * 4);   // reused for both layers
    float* agg1 = (float*)carve((size_t)N * 8 * 4);
    float* agg2 = (float*)carve((size_t)N * 8 * 4);
    float* W1p  = (float*)carve((size_t)F * 16 * 4);
    float* W2p  = (float*)carve((size_t)H * 16 * 4);
    float* gsum = (float*)carve((size_t)G * 8 * 4);

    hipMemsetAsync(deg,  0, (size_t)N * 4,     stream);
    hipMemsetAsync(agg1, 0, (size_t)N * 8 * 4, stream);
    hipMemsetAsync(agg2, 0, (size_t)N * 8 * 4, stream);
    hipMemsetAsync(gsum, 0, (size_t)G * 8 * 4, stream);

    const int tb = 256;
    deg_kernel <<<(E + tb - 1) / tb, tb, 0, stream>>>(dst, deg, E);
    dinv_kernel<<<(N + tb - 1) / tb, tb, 0, stream>>>(deg, dinv, N);
    pad_w_kernel<<<(F * 16 + tb - 1) / tb, tb, 0, stream>>>(W1, W1p, F, H);
    pad_w_kernel<<<(H * 16 + tb - 1) / tb, tb, 0, stream>>>(W2, W2p, H, H);

    int tiles  = (N + 15) / 16;
    int gblk   = (tiles + 7) / 8;   // 8 waves per block

    // Layer 1: h = x @ W1 (WMMA), aggregate, self-loop + bias + ReLU
    wmma_gemm16_kernel<<<gblk, 256, 0, stream>>>(x, F, F, W1p, h, N);
    edge_agg_kernel<<<(E + tb - 1) / tb, tb, 0, stream>>>(src, dst, dinv, h, agg1, E);
    finish_kernel<<<(N * 8 + tb - 1) / tb, tb, 0, stream>>>(h, dinv, b1, agg1, N);

    // Layer 2: h = relu1 @ W2 (WMMA, K=8), aggregate, finish
    wmma_gemm16_kernel<<<gblk, 256, 0, stream>>>(agg1, H, H, W2p, h, N);
    edge_agg_kernel<<<(E + tb - 1) / tb, tb, 0, stream>>>(src, dst, dinv, h, agg2, E);
    finish_kernel<<<(N * 8 + tb - 1) / tb, tb, 0, stream>>>(h, dinv, b2, agg2, N);

    // Readout + sigmoid head
    readout_kernel<<<(N + tb - 1) / tb, tb, 0, stream>>>(batch, agg2, gsum, N);
    logits_kernel<<<(G + tb - 1) / tb, tb, 0, stream>>>(gsum, Wl, bl, (float*)d_out, G);
}